// Multiheadattention_74534862455295
// MI455X (gfx1250) — compile-verified
//
#include <hip/hip_runtime.h>

typedef __attribute__((ext_vector_type(2))) float v2f;
typedef __attribute__((ext_vector_type(4))) float v4f;
typedef __attribute__((ext_vector_type(8))) float v8f;

#define BATCH 2
#define NSEQ  2048
#define CD    768
#define NH    12
#define DH    64
#define BN    (BATCH * NSEQ)   // 4096 rows for the QKV GEMMs

// ---------------------------------------------------------------------------
// CDNA5 async global->LDS copy (no VGPR round trip, tracked by ASYNCcnt).
// LDS offset = low 32 bits of the flat shared-pointer address.
// ---------------------------------------------------------------------------
__device__ __forceinline__ void async_b128(void* lds_ptr, const void* gptr) {
  const unsigned int lds_off = (unsigned int)(uintptr_t)lds_ptr;
  asm volatile("global_load_async_to_lds_b128 %0, %1, off"
               :: "v"(lds_off), "v"(gptr) : "memory");
}
#define WAIT_ASYNC(n) asm volatile("s_wait_asynccnt %0" :: "i"(n) : "memory")

// ---------------------------------------------------------------------------
// QKV projection: y[i,j] = sum_k x[i,k] * w[j,k] + bias[j]
// Block = 128 threads (4 waves). Block tile 64x64, wave tile 16x64.
// K staged 32-wide, double-buffered via async-to-LDS; f32 WMMA 16x16x4.
// ---------------------------------------------------------------------------
__global__ __launch_bounds__(128)
void qkv_gemm_kernel(const float* __restrict__ x, const float* __restrict__ w,
                     const float* __restrict__ bias, float* __restrict__ y) {
  __shared__ float Xs[2][64][36];   // pad 36 -> conflict-free strided frag reads
  __shared__ float Ws[2][64][36];

  const int tid  = threadIdx.x;
  const int lane = tid & 31;
  const int wave = tid >> 5;
  const int lm   = lane & 15;          // M / N index within a 16-wide tile
  const int g    = lane >> 4;          // lane half (0 or 1)
  const int lh   = g << 1;             // K offset contribution: 0 or 2
  const int row0 = blockIdx.x * 64;
  const int col0 = blockIdx.y * 64;

  v8f acc[4];
  for (int t = 0; t < 4; ++t)
    for (int i = 0; i < 8; ++i) acc[t][i] = 0.f;

  // stage one 64x32 slab of X and of W^T into LDS buffer `buf` (8 async ops)
  auto stage = [&](int kc, int buf) {
    const int kc0 = kc * 32;
    for (int i = 0; i < 4; ++i) {
      const int s  = i * 128 + tid;        // 512 float4 slots each
      const int r  = s >> 3;
      const int c4 = (s & 7) << 2;
      async_b128(&Xs[buf][r][c4], &x[(size_t)(row0 + r) * CD + kc0 + c4]);
      async_b128(&Ws[buf][r][c4], &w[(size_t)(col0 + r) * CD + kc0 + c4]);
    }
  };

  stage(0, 0);
  for (int kc = 0; kc < CD / 32; ++kc) {
    const int cur = kc & 1;
    __syncthreads();                       // readers of buf[cur^1] are done
    if (kc + 1 < CD / 32) {
      stage(kc + 1, cur ^ 1);
      WAIT_ASYNC(8);                       // oldest 8 (block kc) landed
    } else {
      WAIT_ASYNC(0);
    }
    __syncthreads();                       // visible to all waves

    for (int kk = 0; kk < 8; ++kk) {
      const int klo = (kk << 2) + lh;
      v2f a;
      a.x = Xs[cur][wave * 16 + lm][klo];
      a.y = Xs[cur][wave * 16 + lm][klo + 1];
      for (int ct = 0; ct < 4; ++ct) {
        v2f bb;                               // B[k, j] = w[j, k]
        bb.x = Ws[cur][ct * 16 + lm][klo];
        bb.y = Ws[cur][ct * 16 + lm][klo + 1];
        acc[ct] = __builtin_amdgcn_wmma_f32_16x16x4_f32(
            false, a, false, bb, (short)0, acc[ct], false, false);
      }
    }
  }

  for (int ct = 0; ct < 4; ++ct) {
    const int j  = col0 + ct * 16 + lm;
    const float bj = bias[j];
    for (int v = 0; v < 8; ++v) {
      const int r = row0 + wave * 16 + v + (g << 3);   // C-layout row mapping
      y[(size_t)r * CD + j] = acc[ct][v] + bj;
    }
  }
}

// ---------------------------------------------------------------------------
// Flash-style attention over the *raw-reshaped* heads.
// Head slab: contiguous [N, 64] at ws + b*N*C + h*N*D.
// Block = 128 threads (4 waves) = one (b, h, 64-query-row block).
// Online softmax; K/V blocks of 64 keys double-buffered via async-to-LDS.
// ---------------------------------------------------------------------------
__global__ __launch_bounds__(128)
void attn_kernel(const float* __restrict__ q_ws, const float* __restrict__ k_ws,
                 const float* __restrict__ v_ws, const int* __restrict__ mask,
                 float* __restrict__ out) {
  __shared__ float Ks[2][64][68];   // 68-stride: conflict-free transposed reads
  __shared__ float Vs[2][64][68];
  __shared__ float Ps[4][16][68];   // per-wave P relayout buffer

  const int tid  = threadIdx.x;
  const int lane = tid & 31;
  const int wave = tid >> 5;
  const int lm   = lane & 15;
  const int g    = lane >> 4;
  const int lh   = g << 1;

  const int qb = blockIdx.x;                 // 0..31 query blocks
  const int bh = blockIdx.y;                 // 0..23
  const int b  = bh / NH;
  const int h  = bh % NH;

  const size_t base = (size_t)b * NSEQ * CD + (size_t)h * NSEQ * DH;
  const float* qp = q_ws + base + (size_t)qb * 64 * DH;
  const float* kp = k_ws + base;
  const float* vp = v_ws + base;

  // Q A-fragments, register resident: 16 rows x 64 K (16 chunks of K=4)
  v2f qf[16];
  const int qrow = wave * 16 + lm;
  for (int c = 0; c < 16; ++c) {
    const float* p = qp + qrow * DH + (c << 2) + lh;
    qf[c].x = p[0];
    qf[c].y = p[1];
  }

  float mrow[8], lrow[8];
  v8f o[4];
  for (int v = 0; v < 8; ++v) { mrow[v] = -3.0e38f; lrow[v] = 0.f; }
  for (int t = 0; t < 4; ++t)
    for (int i = 0; i < 8; ++i) o[t][i] = 0.f;

  const float scale = 3.46410161513775459f;  // sqrt(12): the H**-0.5 quirk

  // stage one 64-key K and V block into LDS buffer `buf` (16 async ops)
  auto stage = [&](int kb, int buf) {
    const float* kblk = kp + kb * 64 * DH;
    const float* vblk = vp + kb * 64 * DH;
    for (int i = 0; i < 8; ++i) {
      const int s  = i * 128 + tid;          // 1024 float4 slots each
      const int r  = s >> 4;
      const int c4 = (s & 15) << 2;
      async_b128(&Ks[buf][r][c4], kblk + r * DH + c4);
      async_b128(&Vs[buf][r][c4], vblk + r * DH + c4);
    }
  };

  stage(0, 0);
  for (int kb = 0; kb < NSEQ / 64; ++kb) {
    const int cur = kb & 1;
    __syncthreads();                         // readers of buf[cur^1] are done
    if (kb + 1 < NSEQ / 64) {
      stage(kb + 1, cur ^ 1);
      WAIT_ASYNC(16);                        // oldest 16 (block kb) landed
    } else {
      WAIT_ASYNC(0);
    }
    if (kb + 2 < NSEQ / 64) {                // warm L2 two blocks ahead
      __builtin_prefetch(kp + (kb + 2) * 64 * DH + tid * 32, 0, 1);
      __builtin_prefetch(vp + (kb + 2) * 64 * DH + tid * 32, 0, 1);
    }
    __syncthreads();                         // block-kb data visible everywhere

    int mk[4];
    for (int nk = 0; nk < 4; ++nk)
      mk[nk] = mask[b * NSEQ + kb * 64 + nk * 16 + lm];

    // S = Q * K^T  (4 tiles of 16 keys, 16 K-chunks each)
    v8f s[4];
    for (int nk = 0; nk < 4; ++nk) {
      for (int i = 0; i < 8; ++i) s[nk][i] = 0.f;
      const int krow = nk * 16 + lm;
      for (int c = 0; c < 16; ++c) {
        const int kcol = (c << 2) + lh;
        v2f bb;                               // B[k, key] = K[key, k]
        bb.x = Ks[cur][krow][kcol];
        bb.y = Ks[cur][krow][kcol + 1];
        s[nk] = __builtin_amdgcn_wmma_f32_16x16x4_f32(
            false, qf[c], false, bb, (short)0, s[nk], false, false);
      }
    }

    // scale, mask-replace, online softmax update
    float alpha[8];
    for (int v = 0; v < 8; ++v) {
      float t = -3.0e38f;
      for (int nk = 0; nk < 4; ++nk) {
        const float val = mk[nk] ? s[nk][v] * scale : -1.0e9f;
        s[nk][v] = val;
        t = fmaxf(t, val);
      }
      for (int off = 1; off < 16; off <<= 1)   // row max across its 16 lanes
        t = fmaxf(t, __shfl_xor(t, off, 32));
      const float mn = fmaxf(mrow[v], t);
      alpha[v] = __expf(mrow[v] - mn);
      mrow[v]  = mn;
      float rs = 0.f;
      for (int nk = 0; nk < 4; ++nk) {
        const float p = __expf(s[nk][v] - mn);
        s[nk][v] = p;
        rs += p;
      }
      for (int off = 1; off < 16; off <<= 1)
        rs += __shfl_xor(rs, off, 32);
      lrow[v] = lrow[v] * alpha[v] + rs;
    }
    for (int t = 0; t < 4; ++t)
      for (int v = 0; v < 8; ++v) o[t][v] *= alpha[v];

    // relayout P: C/D layout -> A-fragment layout via per-wave LDS
    for (int nk = 0; nk < 4; ++nk)
      for (int v = 0; v < 8; ++v)
        Ps[wave][v + (g << 3)][nk * 16 + lm] = s[nk][v];
    __syncthreads();   // uniform; also orders the in-wave LDS RAW

    // O += P * V  (4 d-tiles, 16 key-chunks each)
    for (int dt = 0; dt < 4; ++dt) {
      for (int c = 0; c < 16; ++c) {
        const int pcol = (c << 2) + lh;
        v2f a;
        a.x = Ps[wave][lm][pcol];
        a.y = Ps[wave][lm][pcol + 1];
        const int vrow = (c << 2) + lh;
        const int vcol = dt * 16 + lm;
        v2f bb;                               // B[k, d] = V[key, d]
        bb.x = Vs[cur][vrow][vcol];
        bb.y = Vs[cur][vrow + 1][vcol];
        o[dt] = __builtin_amdgcn_wmma_f32_16x16x4_f32(
            false, a, false, bb, (short)0, o[dt], false, false);
      }
    }
  }

  float inv[8];
  for (int v = 0; v < 8; ++v) inv[v] = 1.f / lrow[v];
  for (int dt = 0; dt < 4; ++dt) {
    const int d = h * DH + dt * 16 + lm;
    for (int v = 0; v < 8; ++v) {
      const int n = qb * 64 + wave * 16 + v + (g << 3);
      out[(size_t)b * NSEQ * CD + (size_t)n * CD + d] = o[dt][v] * inv[v];
    }
  }
}

// ---------------------------------------------------------------------------
extern "C" void kernel_launch(void* const* d_in, const int* in_sizes, int n_in,
                              void* d_out, int out_size, void* d_ws, size_t ws_size,
                              hipStream_t stream) {
  const float* x   = (const float*)d_in[0];
  const int*   msk = (const int*)d_in[1];
  const float* qw  = (const float*)d_in[2];
  const float* qb  = (const float*)d_in[3];
  const float* kw  = (const float*)d_in[4];
  const float* kb  = (const float*)d_in[5];
  const float* vw  = (const float*)d_in[6];
  const float* vb  = (const float*)d_in[7];
  float* out = (float*)d_out;

  const size_t mat = (size_t)BN * CD;     // floats per projection (12.6 MB)
  float* qws = (float*)d_ws;
  float* kws = qws + mat;
  float* vws = kws + mat;

  dim3 gg(BN / 64, CD / 64);              // 64 x 12 blocks
  qkv_gemm_kernel<<<gg, 128, 0, stream>>>(x, qw, qb, qws);
  qkv_gemm_kernel<<<gg, 128, 0, stream>>>(x, kw, kb, kws);
  qkv_gemm_kernel<<<gg, 128, 0, stream>>>(x, vw, vb, vws);

  dim3 ga(NSEQ / 64, BATCH * NH);         // 32 x 24 blocks
  attn_kernel<<<ga, 128, 0, stream>>>(qws, kws, vws, msk, out);
}